// VolumetricFeatureSampler_35845797053053
// MI455X (gfx1250) — compile-verified
//
#include <hip/hip_runtime.h>
#include <hip/hip_bf16.h>

typedef __attribute__((ext_vector_type(2))) float v2f;
typedef __attribute__((ext_vector_type(8))) float v8f;

#define V_ 8
#define F_ 64
#define H_ 256
#define W_ 256
#define G_ 128
#define S_ 512
#define PTS_PER_BLK 256
#define THREADS 256

__global__ __launch_bounds__(THREADS) void vfs_kernel(
    const float* __restrict__ feats,     // (1,V,F,H,W)
    const float* __restrict__ Kmat,      // (1,V,3,3)
    const float* __restrict__ Emat,      // (1,V,3,4)
    const float* __restrict__ dist,      // (1,V,2)
    const float* __restrict__ grids,     // (1,G,3,8,8,8)
    const float* __restrict__ vis,       // (1,V,G)
    const float* __restrict__ normals,   // (1,G,3)
    const float* __restrict__ centers,   // (1,V,3)
    float* __restrict__ out)             // (1,G,2F,8,8,8)
{
    __shared__ float s_px[V_][PTS_PER_BLK];
    __shared__ float s_py[V_][PTS_PER_BLK];
    __shared__ float s_w [V_][PTS_PER_BLK];

    const int blk   = blockIdx.x;           // 0..255
    const int g     = blk >> 1;             // group
    const int sbase = (blk & 1) * PTS_PER_BLK;
    const int tid   = threadIdx.x;
    const int lane  = tid & 31;
    const int waveId = tid >> 5;
    const int row   = lane & 15;
    const bool hi   = lane >= 16;

    const float nx = normals[g*3+0], ny = normals[g*3+1], nz = normals[g*3+2];
    const float* gbase = grids + (size_t)g * 3 * S_;

    // ---------- Phase 1: WMMA-based projection, 16 points x 1 view per task ----------
    const int nTasks = (PTS_PER_BLK / 16) * V_;       // 128
    for (int t = waveId; t < nTasks; t += (THREADS / 32)) {
        const int v      = t >> 4;                    // view 0..7
        const int cchunk = t & 15;                    // 16-point chunk 0..15
        const int lp16   = cchunk * 16;
        const int sl     = sbase + lp16 + row;        // point handled by this "row"

        // xyz for the row's point (all lanes load; needed for B operand + view dirs)
        const float x = gbase[0*S_ + sl];
        const float y = gbase[1*S_ + sl];
        const float z = gbase[2*S_ + sl];

        // A1 = E (3x4) zero-padded to 16x4.  A layout: lanes0-15 -> {K0,K1}, lanes16-31 -> {K2,K3}
        const float* Ev   = Emat + v*12;
        const int   rr    = (row < 3) ? row : 0;
        const float rmask = (row < 3) ? 1.0f : 0.0f;
        v2f a1; v2f b1;
        a1.x = Ev[rr*4 + (hi ? 2 : 0)] * rmask;
        a1.y = Ev[rr*4 + (hi ? 3 : 1)] * rmask;
        // B1 = ph (4x16): VGPR0 lanes0-15 = K0 row (x), lanes16-31 = K2 row (z)
        //                 VGPR1 lanes0-15 = K1 row (y), lanes16-31 = K3 row (1)
        b1.x = hi ? z    : x;
        b1.y = hi ? 1.0f : y;
        v8f acc = {};
        v8f d1 = __builtin_amdgcn_wmma_f32_16x16x4_f32(false, a1, false, b1,
                                                       (short)0, acc, false, false);
        // lanes 0-15: d1[r] = pi[r] for point N=lane (rows 3..15 of A were zero)
        float piz = d1[2];
        float zg  = (fabsf(piz) < 1e-7f) ? 1.0f : piz;
        float rz  = 1.0f / zg;                 // one divide instead of two
        float xp  = d1[0] * rz;
        float yp  = d1[1] * rz;
        float r2  = xp*xp + yp*yp;
        float d0c = dist[v*2+0], d1c = dist[v*2+1];
        float fd  = 1.0f + d0c*r2 + d1c*r2*r2;
        float xd  = xp * fd;
        float yd  = yp * fd;

        // A2 = K (3x3) zero-padded to 16x4;  B2 = [xd; yd; 1; 0]
        const float* Kv = Kmat + v*9;
        v2f a2; v2f b2;
        a2.x = (hi ? Kv[rr*3+2] : Kv[rr*3+0]) * rmask;
        a2.y = hi ? 0.0f : Kv[rr*3+1] * rmask;
        b2.x = hi ? 1.0f : xd;
        b2.y = hi ? 0.0f : yd;
        v8f d2 = __builtin_amdgcn_wmma_f32_16x16x4_f32(false, a2, false, b2,
                                                       (short)0, acc, false, false);

        float u  = 2.0f * d2[0] * (1.0f/255.0f) - 1.0f;
        float vv = 2.0f * d2[1] * (1.0f/255.0f) - 1.0f;
        float inside = (u >= -1.0f && u <= 1.0f && vv >= -1.0f && vv <= 1.0f) ? 1.0f : 0.0f;
        float px = (u  + 1.0f) * (W_ * 0.5f) - 0.5f;
        float py = (vv + 1.0f) * (H_ * 0.5f) - 0.5f;

        // visibility weight: w = softplus_b50( vis * (-min(cos,0)) * inside )
        float cx = centers[v*3+0], cy = centers[v*3+1], cz = centers[v*3+2];
        float dx = x - cx, dy = y - cy, dz = z - cz;
        float nrm = fmaxf(sqrtf(dx*dx + dy*dy + dz*dz), 1e-12f);
        float cosang = fminf((dx*nx + dy*ny + dz*nz) / nrm, 0.0f);
        float wraw = vis[v*G_ + g] * (-cosang) * inside;
        float bx = 50.0f * wraw;
        float w  = (bx > 5.0f) ? wraw : (log1pf(expf(bx)) * (1.0f/50.0f));

        if (!hi) {
            s_px[v][lp16 + row] = px;
            s_py[v][lp16 + row] = py;
            s_w [v][lp16 + row] = w;
        }
    }
    __syncthreads();

    // ---------- Phase 2: bilinear gather + weighted mean/variance over views ----------
    const int lp = tid;                 // local point 0..255
    const int s  = sbase + lp;          // cell within group

    float wn[V_];
    float cw[V_][4];
    int   ci[V_][4];
    float wsum = 0.0f;
    #pragma unroll
    for (int v = 0; v < V_; ++v) { wn[v] = s_w[v][lp]; wsum += wn[v]; }
    const float winv = 1.0f / wsum;     // wsum > 0 always (softplus(>=0) > 0)

    #pragma unroll
    for (int v = 0; v < V_; ++v) {
        wn[v] *= winv;
        float px = s_px[v][lp], py = s_py[v][lp];
        float x0f = floorf(px), y0f = floorf(py);
        float wx = px - x0f, wy = py - y0f;
        int x0 = (int)x0f, y0 = (int)y0f;
        #pragma unroll
        for (int k2 = 0; k2 < 4; ++k2) {
            int dxk = k2 & 1, dyk = k2 >> 1;
            int ix = x0 + dxk, iy = y0 + dyk;
            bool valid = (ix >= 0) && (ix < W_) && (iy >= 0) && (iy < H_);
            int ixc = min(max(ix, 0), W_-1);
            int iyc = min(max(iy, 0), H_-1);
            ci[v][k2] = iyc * W_ + ixc;
            float bw = (dxk ? wx : 1.0f - wx) * (dyk ? wy : 1.0f - wy);
            cw[v][k2] = valid ? bw : 0.0f;
        }
    }

    float* og = out + (size_t)g * (2*F_) * S_ + s;
    #pragma unroll 2
    for (int f = 0; f < F_; ++f) {
        float mean = 0.0f, m2 = 0.0f;
        #pragma unroll
        for (int v = 0; v < V_; ++v) {
            const float* fb = feats + ((size_t)(v*F_ + f) << 16);
            float fv = cw[v][0]*fb[ci[v][0]] + cw[v][1]*fb[ci[v][1]]
                     + cw[v][2]*fb[ci[v][2]] + cw[v][3]*fb[ci[v][3]];
            mean = fmaf(wn[v], fv, mean);
            m2   = fmaf(wn[v]*fv, fv, m2);
        }
        og[(size_t)f * S_]        = mean;
        og[(size_t)(F_ + f) * S_] = m2 - mean*mean;
    }
}

extern "C" void kernel_launch(void* const* d_in, const int* in_sizes, int n_in,
                              void* d_out, int out_size, void* d_ws, size_t ws_size,
                              hipStream_t stream) {
    (void)in_sizes; (void)n_in; (void)out_size; (void)d_ws; (void)ws_size;
    const float* feats   = (const float*)d_in[0];
    const float* Kmat    = (const float*)d_in[1];
    const float* Emat    = (const float*)d_in[2];
    const float* dist    = (const float*)d_in[3];
    const float* grids   = (const float*)d_in[4];
    const float* vis     = (const float*)d_in[5];
    const float* normals = (const float*)d_in[6];
    const float* centers = (const float*)d_in[7];
    float* out = (float*)d_out;

    dim3 grid(G_ * S_ / PTS_PER_BLK);   // 256 blocks
    dim3 block(THREADS);                // 256 threads = 8 wave32
    vfs_kernel<<<grid, block, 0, stream>>>(feats, Kmat, Emat, dist, grids,
                                           vis, normals, centers, out);
}